// AtomicDipolesMACE_38104949850361
// MI455X (gfx1250) — compile-verified
//
#include <hip/hip_runtime.h>
#include <hip/hip_bf16.h>
#include <math.h>

typedef __attribute__((ext_vector_type(2))) float v2f;
typedef __attribute__((ext_vector_type(8))) float v8f;

#define N_NODES 8192
#define N_EDGES 131072
#define FDIM    64
#define NSHC    16
#define NGRAPH  64

__device__ __forceinline__ float silu_f(float x) { return x / (1.0f + __expf(-x)); }

__device__ __forceinline__ v8f wmma4(v2f a, v2f b, v8f c) {
  // D = A(16x4 f32) x B(4x16 f32) + C(16x16 f32)
  return __builtin_amdgcn_wmma_f32_16x16x4_f32(
      /*neg_a=*/false, a, /*neg_b=*/false, b,
      /*c_mod=*/(short)0, c, /*reuse_a=*/false, /*reuse_b=*/false);
}

// ---------------------------------------------------------------------------
// Kernel 1: per-edge geometry + spherical harmonics + radial MLP (4 GEMMs)
// One wave handles 16 edges; WMMA does [16xK]@[Kx64] with silu between stages.
// ---------------------------------------------------------------------------
#define WAVES_A 4
__global__ __launch_bounds__(128) void edge_kernel(
    const float* __restrict__ pos, const float* __restrict__ shifts,
    const int* __restrict__ eidx,
    const float* __restrict__ Wr1, const float* __restrict__ Wr2,
    const float* __restrict__ Wr3, const float* __restrict__ Wr4,
    float* __restrict__ sh_out, float* __restrict__ wedge_out) {
  __shared__ float bufA[WAVES_A][16][66];
  __shared__ float bufB[WAVES_A][16][66];
  const int wave = threadIdx.x >> 5;
  const int lane = threadIdx.x & 31;
  const int half = lane >> 4;
  const int r    = lane & 15;
  const int ebase = (blockIdx.x * WAVES_A + wave) * 16;

  if (lane < 16) {
    const int e  = ebase + lane;
    const int sn = eidx[e];
    const int rn = eidx[N_EDGES + e];
    float vx = pos[rn * 3 + 0] - pos[sn * 3 + 0] + shifts[(size_t)e * 3 + 0];
    float vy = pos[rn * 3 + 1] - pos[sn * 3 + 1] + shifts[(size_t)e * 3 + 1];
    float vz = pos[rn * 3 + 2] - pos[sn * 3 + 2] + shifts[(size_t)e * 3 + 2];
    float len = sqrtf(vx * vx + vy * vy + vz * vz);
    float inv = 1.0f / (len + 1e-9f);
    float x = vx * inv, y = vy * inv, z = vz * inv;
    const float s3 = 1.7320508075688772f, s5 = 2.2360679774997896f;
    const float s15 = 3.872983346207417f, s7 = 2.6457513110645907f;
    const float s105 = 10.246950765959598f;
    const float s35_8 = 2.091650066335189f, s21_8 = 1.620185174601965f;
    float* shp = sh_out + (size_t)e * NSHC;
    shp[0]  = 1.0f;
    shp[1]  = s3 * x;  shp[2] = s3 * y;  shp[3] = s3 * z;
    shp[4]  = s15 * x * y;
    shp[5]  = s15 * y * z;
    shp[6]  = 0.5f * s5 * (3.0f * z * z - 1.0f);
    shp[7]  = s15 * x * z;
    shp[8]  = 0.5f * s15 * (x * x - y * y);
    shp[9]  = s35_8 * y * (3.0f * x * x - y * y);
    shp[10] = s105 * x * y * z;
    shp[11] = s21_8 * y * (5.0f * z * z - 1.0f);
    shp[12] = 0.5f * s7 * (5.0f * z * z * z - 3.0f * z);
    shp[13] = s21_8 * x * (5.0f * z * z - 1.0f);
    shp[14] = 0.5f * s105 * z * (x * x - y * y);
    shp[15] = s35_8 * x * (3.0f * x * x - y * y);
    // radial embedding: bessel * polynomial cutoff (p=5, R_MAX=5)
    float u = len * 0.2f;
    float u2 = u * u;
    float u5 = u2 * u2 * u;
    float fc = 1.0f - 21.0f * u5 + 35.0f * u5 * u - 15.0f * u5 * u2;
    fc = (u < 1.0f) ? fc : 0.0f;
    float pref = 0.6324555320336759f * inv * fc;  // sqrt(2/5)/(r+1e-9)*fc
    const float pi_over_rmax = 0.6283185307179586f;
#pragma unroll
    for (int n = 1; n <= 8; n++)
      bufA[wave][lane][n - 1] = pref * __sinf((float)n * pi_over_rmax * len);
  }
  __syncthreads();

  v8f acc[4];
  // ---- Stage 1: rb[16x8] @ Wr1[8x64] -> silu -> bufB
#pragma unroll
  for (int nt = 0; nt < 4; nt++) acc[nt] = {};
  for (int k0 = 0; k0 < 8; k0 += 4) {
    int kk = k0 + half * 2;
    v2f a; a.x = bufA[wave][r][kk]; a.y = bufA[wave][r][kk + 1];
#pragma unroll
    for (int nt = 0; nt < 4; nt++) {
      int col = nt * 16 + r;
      v2f b; b.x = Wr1[kk * 64 + col]; b.y = Wr1[(kk + 1) * 64 + col];
      acc[nt] = wmma4(a, b, acc[nt]);
    }
  }
#pragma unroll
  for (int nt = 0; nt < 4; nt++)
#pragma unroll
    for (int i = 0; i < 8; i++)
      bufB[wave][i + half * 8][nt * 16 + r] = silu_f(acc[nt][i]);

  // ---- Stage 2: bufB[16x64] @ Wr2 -> silu -> bufA
#pragma unroll
  for (int nt = 0; nt < 4; nt++) acc[nt] = {};
  for (int k0 = 0; k0 < 64; k0 += 4) {
    int kk = k0 + half * 2;
    v2f a; a.x = bufB[wave][r][kk]; a.y = bufB[wave][r][kk + 1];
#pragma unroll
    for (int nt = 0; nt < 4; nt++) {
      int col = nt * 16 + r;
      v2f b; b.x = Wr2[kk * 64 + col]; b.y = Wr2[(kk + 1) * 64 + col];
      acc[nt] = wmma4(a, b, acc[nt]);
    }
  }
#pragma unroll
  for (int nt = 0; nt < 4; nt++)
#pragma unroll
    for (int i = 0; i < 8; i++)
      bufA[wave][i + half * 8][nt * 16 + r] = silu_f(acc[nt][i]);

  // ---- Stage 3: bufA[16x64] @ Wr3 -> silu -> bufB
#pragma unroll
  for (int nt = 0; nt < 4; nt++) acc[nt] = {};
  for (int k0 = 0; k0 < 64; k0 += 4) {
    int kk = k0 + half * 2;
    v2f a; a.x = bufA[wave][r][kk]; a.y = bufA[wave][r][kk + 1];
#pragma unroll
    for (int nt = 0; nt < 4; nt++) {
      int col = nt * 16 + r;
      v2f b; b.x = Wr3[kk * 64 + col]; b.y = Wr3[(kk + 1) * 64 + col];
      acc[nt] = wmma4(a, b, acc[nt]);
    }
  }
#pragma unroll
  for (int nt = 0; nt < 4; nt++)
#pragma unroll
    for (int i = 0; i < 8; i++)
      bufB[wave][i + half * 8][nt * 16 + r] = silu_f(acc[nt][i]);

  // ---- Stage 4: bufB[16x64] @ Wr4 -> w_edge (no activation)
#pragma unroll
  for (int nt = 0; nt < 4; nt++) acc[nt] = {};
  for (int k0 = 0; k0 < 64; k0 += 4) {
    int kk = k0 + half * 2;
    v2f a; a.x = bufB[wave][r][kk]; a.y = bufB[wave][r][kk + 1];
#pragma unroll
    for (int nt = 0; nt < 4; nt++) {
      int col = nt * 16 + r;
      v2f b; b.x = Wr4[kk * 64 + col]; b.y = Wr4[(kk + 1) * 64 + col];
      acc[nt] = wmma4(a, b, acc[nt]);
    }
  }
#pragma unroll
  for (int nt = 0; nt < 4; nt++)
#pragma unroll
    for (int i = 0; i < 8; i++)
      wedge_out[(size_t)(ebase + i + half * 8) * 64 + nt * 16 + r] = acc[nt][i];
}

// ---------------------------------------------------------------------------
// Kernel 2: node embedding -> feats m=0 plane (feats pre-zeroed by memset)
// ---------------------------------------------------------------------------
__global__ __launch_bounds__(256) void embed_kernel(
    const float* __restrict__ attrs, const float* __restrict__ Wemb,
    float* __restrict__ feats) {
  int t = blockIdx.x * 256 + threadIdx.x;
  int n = t >> 6, f = t & 63;
  float acc = 0.0f;
#pragma unroll
  for (int k = 0; k < 10; k++) acc += attrs[n * 10 + k] * Wemb[k * 64 + f];
  feats[(size_t)n * 64 + f] = acc;  // m = 0 plane of [m][n][f]
}

// ---------------------------------------------------------------------------
// Kernel 3: hh[m] = feats[m] @ W_up   (16 GEMMs of [N,64]@[64,64], WMMA)
// One wave per (m, 16-node tile).
// ---------------------------------------------------------------------------
__global__ __launch_bounds__(256) void up_kernel(
    const float* __restrict__ feats, const float* __restrict__ W,
    float* __restrict__ hh) {
  const int wid  = blockIdx.x * 8 + (threadIdx.x >> 5);
  const int m    = wid >> 9;      // N/16 = 512 tiles per m
  const int tile = wid & 511;
  const int lane = threadIdx.x & 31;
  const int half = lane >> 4, r = lane & 15;
  const float* A = feats + ((size_t)m * N_NODES + (size_t)tile * 16) * 64;
  v8f acc[4];
#pragma unroll
  for (int nt = 0; nt < 4; nt++) acc[nt] = {};
  for (int k0 = 0; k0 < 64; k0 += 4) {
    int kk = k0 + half * 2;
    v2f a; a.x = A[r * 64 + kk]; a.y = A[r * 64 + kk + 1];
#pragma unroll
    for (int nt = 0; nt < 4; nt++) {
      int col = nt * 16 + r;
      v2f b; b.x = W[kk * 64 + col]; b.y = W[(kk + 1) * 64 + col];
      acc[nt] = wmma4(a, b, acc[nt]);
    }
  }
  float* D = hh + ((size_t)m * N_NODES + (size_t)tile * 16) * 64;
#pragma unroll
  for (int nt = 0; nt < 4; nt++)
#pragma unroll
    for (int i = 0; i < 8; i++)
      D[(i + half * 8) * 64 + nt * 16 + r] = acc[nt][i];
}

// ---------------------------------------------------------------------------
// Kernel 4: message construction + scatter-add (segment_sum via f32 atomics)
// 64 threads per edge (one per feature f); each handles all 16 m.
// ---------------------------------------------------------------------------
__global__ __launch_bounds__(256) void msg_kernel(
    const int* __restrict__ eidx, const float* __restrict__ wedge,
    const float* __restrict__ sh, const float* __restrict__ hh,
    float* __restrict__ agg) {
  int t = blockIdx.x * 256 + threadIdx.x;
  int e = t >> 6, f = t & 63;
  int sn = eidx[e];
  int rn = eidx[N_EDGES + e];
  float we  = wedge[(size_t)e * 64 + f];
  float sh0 = sh[(size_t)e * NSHC];
  float hs0 = hh[(size_t)sn * 64 + f];  // m = 0
#pragma unroll
  for (int m = 0; m < NSHC; m++) {
    float shm = sh[(size_t)e * NSHC + m];
    float hsm = hh[((size_t)m * N_NODES + sn) * 64 + f];
    float msg = we * (hs0 * shm + hsm * sh0);
    unsafeAtomicAdd(&agg[((size_t)m * N_NODES + rn) * 64 + f], msg);
  }
}

// ---------------------------------------------------------------------------
// Kernel 5: s2[n,f] = node_attrs[n] @ W_sc[:,f]
// ---------------------------------------------------------------------------
__global__ __launch_bounds__(256) void sc_kernel(
    const float* __restrict__ attrs, const float* __restrict__ Wsc,
    float* __restrict__ s2) {
  int t = blockIdx.x * 256 + threadIdx.x;
  int n = t >> 6, f = t & 63;
  float acc = 0.0f;
#pragma unroll
  for (int k = 0; k < 10; k++) acc += attrs[n * 10 + k] * Wsc[k * 64 + f];
  s2[(size_t)n * 64 + f] = acc;
}

// ---------------------------------------------------------------------------
// Kernel 6: fnew[m] = (agg[m]/16) @ W_mix + feats_old[m]*s2  (WMMA + epilogue)
// ---------------------------------------------------------------------------
__global__ __launch_bounds__(256) void mix_kernel(
    const float* __restrict__ agg, const float* __restrict__ W,
    const float* __restrict__ fold, const float* __restrict__ s2,
    float* __restrict__ fnew) {
  const int wid  = blockIdx.x * 8 + (threadIdx.x >> 5);
  const int m    = wid >> 9;
  const int tile = wid & 511;
  const int lane = threadIdx.x & 31;
  const int half = lane >> 4, r = lane & 15;
  const float* A = agg + ((size_t)m * N_NODES + (size_t)tile * 16) * 64;
  v8f acc[4];
#pragma unroll
  for (int nt = 0; nt < 4; nt++) acc[nt] = {};
  for (int k0 = 0; k0 < 64; k0 += 4) {
    int kk = k0 + half * 2;
    v2f a;
    a.x = A[r * 64 + kk] * 0.0625f;       // / AVG_NEIGH
    a.y = A[r * 64 + kk + 1] * 0.0625f;
#pragma unroll
    for (int nt = 0; nt < 4; nt++) {
      int col = nt * 16 + r;
      v2f b; b.x = W[kk * 64 + col]; b.y = W[(kk + 1) * 64 + col];
      acc[nt] = wmma4(a, b, acc[nt]);
    }
  }
#pragma unroll
  for (int nt = 0; nt < 4; nt++)
#pragma unroll
    for (int i = 0; i < 8; i++) {
      int row = tile * 16 + i + half * 8;
      int col = nt * 16 + r;
      size_t pidx = ((size_t)m * N_NODES + row) * 64 + col;
      fnew[pidx] = acc[nt][i] + fold[pidx] * s2[(size_t)row * 64 + col];
    }
}

// ---------------------------------------------------------------------------
// Kernel 7: gated nonlinearity (in place) + dipole readout accumulation
// ---------------------------------------------------------------------------
__global__ __launch_bounds__(256) void nonlin_kernel(
    float* __restrict__ feats, const float* __restrict__ wread,
    float* __restrict__ dip) {
  int t = blockIdx.x * 256 + threadIdx.x;
  int n = t >> 6, f = t & 63;
  float t0 = feats[(size_t)n * 64 + f];     // m = 0 (post mixed+sc)
  float scale = 1.0f + 0.5f * t0;
  float wr = wread[f];
  float d[3] = {0.0f, 0.0f, 0.0f};
#pragma unroll
  for (int m = 0; m < NSHC; m++) {
    size_t idx = ((size_t)m * N_NODES + n) * 64 + f;
    float v = feats[idx] * scale;
    feats[idx] = v;
    if (m >= 1 && m <= 3) d[m - 1] = v;
  }
#pragma unroll
  for (int c = 0; c < 3; c++)
    unsafeAtomicAdd(&dip[(size_t)n * 3 + c], wr * d[c]);
}

// ---------------------------------------------------------------------------
// Kernel 8: graph-level segment sums (total = seg(dip) + seg(charge*pos))
// ---------------------------------------------------------------------------
__global__ __launch_bounds__(256) void final_kernel(
    const float* __restrict__ dip, const float* __restrict__ charges,
    const float* __restrict__ pos, const int* __restrict__ batch,
    float* __restrict__ total) {
  int n = blockIdx.x * 256 + threadIdx.x;
  if (n >= N_NODES) return;
  int g = batch[n];
  float q = charges[n];
#pragma unroll
  for (int c = 0; c < 3; c++)
    unsafeAtomicAdd(&total[g * 3 + c],
                    dip[(size_t)n * 3 + c] + q * pos[(size_t)n * 3 + c]);
}

// ---------------------------------------------------------------------------
extern "C" void kernel_launch(void* const* d_in, const int* in_sizes, int n_in,
                              void* d_out, int out_size, void* d_ws, size_t ws_size,
                              hipStream_t stream) {
  const float* positions  = (const float*)d_in[0];
  const float* node_attrs = (const float*)d_in[1];
  const float* shifts     = (const float*)d_in[2];
  const float* charges    = (const float*)d_in[3];
  const float* W_emb      = (const float*)d_in[4];
  const float* W_r1       = (const float*)d_in[5];
  const float* W_r2       = (const float*)d_in[6];
  const float* W_r3       = (const float*)d_in[7];
  const float* W_r4       = (const float*)d_in[8];
  const float* W_up       = (const float*)d_in[9];
  const float* W_mix      = (const float*)d_in[10];
  const float* W_sc       = (const float*)d_in[11];
  const float* w_read     = (const float*)d_in[12];
  const int*   edge_index = (const int*)d_in[13];
  const int*   batch      = (const int*)d_in[14];
  float* out = (float*)d_out;
  float* ws  = (float*)d_ws;

  const size_t PLANE = (size_t)NSHC * N_NODES * FDIM;  // 8.39M floats
  size_t off = 0;
  float* sh_buf = ws + off;  off += (size_t)N_EDGES * NSHC;
  float* wedge  = ws + off;  off += (size_t)N_EDGES * FDIM;
  float* featsA = ws + off;  off += PLANE;
  float* featsB = ws + off;  off += PLANE;
  float* hh     = ws + off;  off += PLANE;
  float* agg    = ws + off;  off += PLANE;
  float* s2     = ws + off;  off += (size_t)N_NODES * FDIM;
  (void)in_sizes; (void)n_in; (void)ws_size; (void)out_size;

  float* dip = out + NGRAPH * 3;  // second tuple element lives after total

  // zero output (total + atomic_dipoles accumulators) and feats m>0 planes
  hipMemsetAsync(d_out, 0, (size_t)(NGRAPH * 3 + N_NODES * 3) * sizeof(float), stream);
  hipMemsetAsync(featsA, 0, PLANE * sizeof(float), stream);

  // edge geometry + SH + radial MLP
  edge_kernel<<<N_EDGES / (16 * WAVES_A), 128, 0, stream>>>(
      positions, shifts, edge_index, W_r1, W_r2, W_r3, W_r4, sh_buf, wedge);

  // node embedding (m=0 plane)
  embed_kernel<<<(N_NODES * FDIM) / 256, 256, 0, stream>>>(node_attrs, W_emb, featsA);

  float* cur = featsA;
  float* nxt = featsB;
  for (int layer = 0; layer < 2; layer++) {
    up_kernel<<<(NSHC * (N_NODES / 16)) / 8, 256, 0, stream>>>(
        cur, W_up + (size_t)layer * 64 * 64, hh);
    hipMemsetAsync(agg, 0, PLANE * sizeof(float), stream);
    msg_kernel<<<((size_t)N_EDGES * FDIM) / 256, 256, 0, stream>>>(
        edge_index, wedge, sh_buf, hh, agg);
    sc_kernel<<<(N_NODES * FDIM) / 256, 256, 0, stream>>>(
        node_attrs, W_sc + (size_t)layer * 10 * 64, s2);
    mix_kernel<<<(NSHC * (N_NODES / 16)) / 8, 256, 0, stream>>>(
        agg, W_mix + (size_t)layer * 64 * 64, cur, s2, nxt);
    nonlin_kernel<<<(N_NODES * FDIM) / 256, 256, 0, stream>>>(
        nxt, w_read + (size_t)layer * 64, dip);
    float* tmp = cur; cur = nxt; nxt = tmp;
  }

  final_kernel<<<(N_NODES + 255) / 256, 256, 0, stream>>>(
      dip, charges, positions, batch, out);
}